// GCN_41128606826857
// MI455X (gfx1250) — compile-verified
//
#include <hip/hip_runtime.h>
#include <stdint.h>

// ---------------------------------------------------------------------------
// 3-layer GCN for MI455X (gfx1250, wave32).
// - GEMMs: V_WMMA_F32_16X16X4_F32, one wave per 16-row M-tile computing ALL
//   F-tiles (A fragment loaded once, 6x less X traffic than tile-per-wave).
// - W staged in LDS once per block; B fragments read via ds_load_b64.
// - BN(eval)+ReLU fused into the A-fragment load of the consuming GEMM
//   (exact: BN'd h is only ever consumed through h @ W^T).
// - Edge scatter (the ~2 GB roofline bottleneck) via coalesced float4 gather
//   + global_atomic_add_f32.
// ---------------------------------------------------------------------------

#define IN_F   128
#define HID    96
#define OUT_F  64
#define BN_EPS 1e-5f

typedef __attribute__((ext_vector_type(2))) float v2f;
typedef __attribute__((ext_vector_type(8))) float v8f;

// ---------------------------------------------------------------------------
// Degree / norm kernels
// ---------------------------------------------------------------------------
__global__ void gcn_deg_init(float* __restrict__ deg, int n) {
    int i = blockIdx.x * blockDim.x + threadIdx.x;
    if (i < n) deg[i] = 1.0f;                 // self-loop weight
}

__global__ void gcn_deg_accum(const long long* __restrict__ col,
                              const float* __restrict__ ew,
                              float* __restrict__ deg, int e) {
    int i = blockIdx.x * blockDim.x + threadIdx.x;
    if (i < e) atomicAdd(&deg[(int)col[i]], ew[i]);
}

__global__ void gcn_dinv(const float* __restrict__ deg,
                         float* __restrict__ dinv, int n) {
    int i = blockIdx.x * blockDim.x + threadIdx.x;
    if (i < n) {
        float d = deg[i];
        dinv[i] = (d > 0.0f) ? rsqrtf(d) : 0.0f;
    }
}

// scale = g * rsqrt(v + eps); shift = beta - m * scale
__global__ void gcn_bn_coeff(const float* __restrict__ g,
                             const float* __restrict__ be,
                             const float* __restrict__ m,
                             const float* __restrict__ v,
                             float* __restrict__ scale,
                             float* __restrict__ shift, int f) {
    int i = blockIdx.x * blockDim.x + threadIdx.x;
    if (i < f) {
        float s = g[i] * rsqrtf(v[i] + BN_EPS);
        scale[i] = s;
        shift[i] = be[i] - m[i] * s;
    }
}

// ---------------------------------------------------------------------------
// WMMA GEMM: C[n, F] = act(X)[n, K] @ W[F, K]^T
// One wave handles one 16-row M-tile and ALL F/16 F-tiles.
// act = identity (BN=false) or ReLU(BN affine) fused on the A-fragment load.
//
// A-frag (16x4 f32, ISA layout): lanes 0-15 -> (K, K+1); lanes 16-31 ->
// (K+2, K+3): contiguous float2 per lane. B-frag mirrors it over W rows.
// C/D: VGPR r -> (M = r + 8*half, N = lane&15).
// ---------------------------------------------------------------------------
template <int K, int F, bool BN>
__global__ void gcn_gemm_wmma(const float* __restrict__ X,
                              const float* __restrict__ W,
                              float* __restrict__ C, int nrows,
                              const float* __restrict__ bnscale,
                              const float* __restrict__ bnshift) {
    constexpr int FT = F / 16;
    __shared__ float sW[K * F];
    __shared__ float sScale[BN ? K : 1];
    __shared__ float sShift[BN ? K : 1];

    for (int i = threadIdx.x; i < K * F; i += blockDim.x) sW[i] = W[i];
    if (BN) {
        for (int i = threadIdx.x; i < K; i += blockDim.x) {
            sScale[i] = bnscale[i];
            sShift[i] = bnshift[i];
        }
    }
    __syncthreads();

    const int lane  = threadIdx.x & 31;
    const int wave  = threadIdx.x >> 5;
    const int mtile = blockIdx.x * (blockDim.x >> 5) + wave;
    if (mtile * 16 >= nrows) return;          // wave-uniform: EXEC all-1s

    const int m0   = mtile * 16;
    const int half = lane >> 4;               // 0: lanes 0-15, 1: lanes 16-31
    const int l16  = lane & 15;

    const float* xrow = X + (size_t)(m0 + l16) * K;

    v8f acc[FT];
    #pragma unroll
    for (int ft = 0; ft < FT; ++ft) acc[ft] = (v8f){};

    #pragma unroll 4
    for (int k = 0; k < K; k += 4) {
        const int ka = k + 2 * half;
        v2f a = *(const v2f*)(xrow + ka);
        if (BN) {
            a.x = fmaxf(fmaf(a.x, sScale[ka],     sShift[ka]),     0.0f);
            a.y = fmaxf(fmaf(a.y, sScale[ka + 1], sShift[ka + 1]), 0.0f);
        }
        #pragma unroll
        for (int ft = 0; ft < FT; ++ft) {
            v2f b = *(const v2f*)(sW + (size_t)(ft * 16 + l16) * K + ka);
            acc[ft] = __builtin_amdgcn_wmma_f32_16x16x4_f32(
                /*neg_a=*/false, a, /*neg_b=*/false, b,
                /*c_mod=*/(short)0, acc[ft], /*reuse_a=*/false, /*reuse_b=*/false);
        }
    }

    #pragma unroll
    for (int ft = 0; ft < FT; ++ft) {
        float* crow = C + (size_t)(m0 + 8 * half) * F + ft * 16 + l16;
        #pragma unroll
        for (int r = 0; r < 8; ++r)
            crow[(size_t)r * F] = acc[ft][r];
    }
}

// ---------------------------------------------------------------------------
// out[i, f] = dinv[i]^2 * xw[i, f] + bias[f]   (self-loop term; writes every
// element, fully initializing the poisoned destination)
// ---------------------------------------------------------------------------
template <int F>
__global__ void gcn_self_bias(const float* __restrict__ xw,
                              const float* __restrict__ dinv,
                              const float* __restrict__ bias,
                              float* __restrict__ out, int n) {
    int tid = blockIdx.x * blockDim.x + threadIdx.x;
    if (tid >= n * F) return;
    int i = tid / F;
    int f = tid - i * F;
    float d = dinv[i];
    out[tid] = d * d * xw[tid] + bias[f];
}

// ---------------------------------------------------------------------------
// Edge scatter: agg[col] += (dinv[row]*ew*dinv[col]) * xw[row]
// One thread per (edge, 4-feature chunk): consecutive threads share the edge
// and walk consecutive chunks -> coalesced loads and atomics.
// ---------------------------------------------------------------------------
template <int F>
__global__ void gcn_edge_scatter(const long long* __restrict__ rowi,
                                 const long long* __restrict__ coli,
                                 const float* __restrict__ ew,
                                 const float* __restrict__ dinv,
                                 const float* __restrict__ xw,
                                 float* __restrict__ agg, int e) {
    constexpr int F4 = F / 4;
    int tid  = blockIdx.x * blockDim.x + threadIdx.x;
    int edge = tid / F4;
    int c    = tid - edge * F4;
    if (edge >= e) return;
    int r  = (int)rowi[edge];
    int cl = (int)coli[edge];
    float nrm = dinv[r] * ew[edge] * dinv[cl];
    const float4 v = *(const float4*)(xw + (size_t)r * F + 4 * c);
    float* dst = agg + (size_t)cl * F + 4 * c;
    atomicAdd(dst + 0, nrm * v.x);
    atomicAdd(dst + 1, nrm * v.y);
    atomicAdd(dst + 2, nrm * v.z);
    atomicAdd(dst + 3, nrm * v.w);
}

// ---------------------------------------------------------------------------
// Host-side launcher (graph-capture safe: only kernel launches on `stream`)
// ---------------------------------------------------------------------------
extern "C" void kernel_launch(void* const* d_in, const int* in_sizes, int n_in,
                              void* d_out, int out_size, void* d_ws, size_t ws_size,
                              hipStream_t stream) {
    const float*     x   = (const float*)d_in[0];
    const long long* ei  = (const long long*)d_in[1];
    const float*     ew  = (const float*)d_in[2];
    const float*     W1  = (const float*)d_in[3];
    const float*     b1  = (const float*)d_in[4];
    const float*     W2  = (const float*)d_in[5];
    const float*     b2  = (const float*)d_in[6];
    const float*     W3  = (const float*)d_in[7];
    const float*     b3  = (const float*)d_in[8];
    const float*     g1  = (const float*)d_in[9];
    const float*     be1 = (const float*)d_in[10];
    const float*     m1  = (const float*)d_in[11];
    const float*     v1  = (const float*)d_in[12];
    const float*     g2  = (const float*)d_in[13];
    const float*     be2 = (const float*)d_in[14];
    const float*     m2  = (const float*)d_in[15];
    const float*     v2  = (const float*)d_in[16];

    const int n = in_sizes[0] / IN_F;      // 50000
    const int e = in_sizes[2];             // 800000
    const long long* row = ei;             // edge_index[0] = sources
    const long long* col = ei + e;         // edge_index[1] = targets

    // Workspace (floats):
    //   xw[n*HID] | agg[n*HID] | deg[n] | dinv[n] | sc1[HID] sh1[HID] sc2[HID] sh2[HID]
    float* ws   = (float*)d_ws;
    float* xw   = ws;
    float* agg  = ws + (size_t)n * HID;
    float* deg  = ws + (size_t)2 * n * HID;
    float* dinv = deg + n;
    float* sc1  = dinv + n;
    float* sh1  = sc1 + HID;
    float* sc2  = sh1 + HID;
    float* sh2  = sc2 + HID;
    float* out  = (float*)d_out;

    const int T = 256;
    auto cdiv = [](long long a, long long b) { return (int)((a + b - 1) / b); };

    // --- GCN normalization + BN coefficients ---
    gcn_deg_init<<<cdiv(n, T), T, 0, stream>>>(deg, n);
    gcn_deg_accum<<<cdiv(e, T), T, 0, stream>>>(col, ew, deg, e);
    gcn_dinv<<<cdiv(n, T), T, 0, stream>>>(deg, dinv, n);
    gcn_bn_coeff<<<1, HID, 0, stream>>>(g1, be1, m1, v1, sc1, sh1, HID);
    gcn_bn_coeff<<<1, HID, 0, stream>>>(g2, be2, m2, v2, sc2, sh2, HID);

    const int mtiles = cdiv(n, 16);
    const int gblocks = cdiv(mtiles, 4);   // 4 waves (M-tiles) per block

    // --- Layer 1: x @ W1^T (no input activation) ---
    gcn_gemm_wmma<IN_F, HID, false><<<gblocks, 128, 0, stream>>>(
        x, W1, xw, n, nullptr, nullptr);
    gcn_self_bias<HID><<<cdiv((long long)n * HID, T), T, 0, stream>>>(xw, dinv, b1, agg, n);
    gcn_edge_scatter<HID><<<cdiv((long long)e * (HID / 4), T), T, 0, stream>>>(
        row, col, ew, dinv, xw, agg, e);
    // agg now holds pre-BN h1

    // --- Layer 2: relu(BN1(h1)) @ W2^T (BN+ReLU fused into A load) ---
    gcn_gemm_wmma<HID, HID, true><<<gblocks, 128, 0, stream>>>(
        agg, W2, xw, n, sc1, sh1);
    gcn_self_bias<HID><<<cdiv((long long)n * HID, T), T, 0, stream>>>(xw, dinv, b2, agg, n);
    gcn_edge_scatter<HID><<<cdiv((long long)e * (HID / 4), T), T, 0, stream>>>(
        row, col, ew, dinv, xw, agg, e);
    // agg now holds pre-BN h2

    // --- Layer 3: relu(BN2(h2)) @ W3^T, no output BN ---
    gcn_gemm_wmma<HID, OUT_F, true><<<gblocks, 128, 0, stream>>>(
        agg, W3, xw, n, sc2, sh2);
    gcn_self_bias<OUT_F><<<cdiv((long long)n * OUT_F, T), T, 0, stream>>>(xw, dinv, b3, out, n);
    gcn_edge_scatter<OUT_F><<<cdiv((long long)e * (OUT_F / 4), T), T, 0, stream>>>(
        row, col, ew, dinv, xw, out, e);
}